// TensorEncoder_77721728188948
// MI455X (gfx1250) — compile-verified
//
#include <hip/hip_runtime.h>

#define RES   300
#define NC    48
#define NPTS  1000000
#define CELLS (RES * RES)              // 90000 cells per plane
#define PLANE_F (3 * CELLS * NC)       // 12,960,000 floats (all 3 transposed planes)
#define C4    (NC / 4)                 // 12 float4 groups per 48-channel cell

typedef float v4f __attribute__((ext_vector_type(4)));
typedef unsigned int v4u __attribute__((ext_vector_type(4)));
typedef int v4i __attribute__((ext_vector_type(4)));
typedef int v8i __attribute__((ext_vector_type(8)));

// ---------------------------------------------------------------------------
// Pre-pass 1 (TDM version): (3, 48, 300, 300) -> (3, 300, 300, 48)
// One block per (y, s). The Tensor Data Mover DMAs a strided 2D tile
// (tile_dim0 = 300 contiguous floats along x, tile_dim1 = 48 channel rows,
// row stride = 90000 floats) from the channel-major layout into LDS with a
// single tensor_load_to_lds. The block then writes the transposed tile back
// out as fully contiguous 14400-float runs.
// ---------------------------------------------------------------------------
__global__ void __launch_bounds__(288)
tpose_planes_tdm(const float* __restrict__ in, float* __restrict__ out) {
    __shared__ float tile[NC * RES];                 // 57,600 B: tile[c*300 + x]
    const int y = blockIdx.x;                        // 0..299
    const int s = blockIdx.y;                        // 0..2
    const int tid = threadIdx.x + threadIdx.y * 48;  // block (48,6)

    if (tid < 32) {                                  // exactly wave 0 issues the DMA
        unsigned long long ga =
            (unsigned long long)(const void*)(in + (s * NC) * CELLS + y * RES);
        unsigned int lds = (unsigned int)(unsigned long long)&tile[0]; // LDS byte offset

        // D# group 0: count=1 | lds_addr | global_addr[56:0] | type=2
        v4u g0;
        g0.x = 0x1u;
        g0.y = lds;
        g0.z = (unsigned int)(ga & 0xffffffffu);
        g0.w = (unsigned int)((ga >> 32) & 0x1ffffffu) | (2u << 30);

        // D# group 1: data_size=4B; tensor 300x48; tile 300x48; dim0 stride 90000
        v8i g1;
        g1[0] = 0x00020000;                  // workgroup_mask=0, data_size=2 (4B)
        g1[1] = (RES & 0xffff) << 16;        // tensor_dim0[15:0]   = 300
        g1[2] = (NC & 0xffff) << 16;         // tensor_dim0 hi=0 | tensor_dim1 lo = 48
        g1[3] = (RES & 0xffff) << 16;        // tensor_dim1 hi=0 | tile_dim0 = 300
        g1[4] = NC;                          // tile_dim1 = 48, tile_dim2 = 0 (2D)
        g1[5] = CELLS;                       // tensor_dim0_stride = 90000 (lo)
        g1[6] = 0;                           // stride hi = 0, tensor_dim1_stride lo = 0
        g1[7] = 0;

        v4i g2 = {0, 0, 0, 0};               // 2D tile: dims 2..4 unused
        v4i g3 = {0, 0, 0, 0};
#if __clang_major__ >= 23
        v8i g4 = {0, 0, 0, 0, 0, 0, 0, 0};
        __builtin_amdgcn_tensor_load_to_lds(g0, g1, g2, g3, g4, 0);
#else
        __builtin_amdgcn_tensor_load_to_lds(g0, g1, g2, g3, 0);
#endif
        __builtin_amdgcn_s_wait_tensorcnt(0);
    }
    __syncthreads();

    // out[((s*CELLS + y*300 + x) * 48) + c]: one contiguous 57.6KB span per block.
    float* dst = out + (s * CELLS + y * RES) * NC;
    const int c = threadIdx.x;                       // 0..47: lane-contiguous stores
#pragma unroll 5
    for (int k = 0; k < 50; ++k) {
        const int x = k * 6 + threadIdx.y;
        dst[x * NC + c] = tile[c * RES + x];
    }
}

// Pre-pass 2: (3, 48, 300, 1) -> (3, 300, 48)   (tiny; plain version)
__global__ void tpose_lines(const float* __restrict__ in, float* __restrict__ out) {
    const int y  = threadIdx.x;            // blockDim.x = 320, 300 active
    if (y >= RES) return;
    const int sc = blockIdx.y;             // 0..143
    const int s  = sc / NC;
    const int c  = sc - s * NC;
    float v = __builtin_nontemporal_load(&in[sc * RES + y]);
    out[(s * RES + y) * NC + c] = v;
}

// align_corners grid-sample coord: px = (g+1)*0.5*(R-1) = fma(g, 149.5, 149.5)
__device__ __forceinline__ void coord(float g, int& i0, int& i1, float& w) {
    float p = fmaf(g, 0.5f * (RES - 1), 0.5f * (RES - 1));
    float f = floorf(p);
    w = p - f;
    int i = (int)f;
    i0 = min(max(i, 0), RES - 1);
    i1 = min(i0 + 1, RES - 1);
}

// ---------------------------------------------------------------------------
// Main gather kernel. block = (6, 32): 6 lanes x two float4 groups = 8
// channels/thread, 32 points. Coalesced b128 gathers from the L2-resident
// transposed planes; non-temporal b128 stores for the 576MB output stream.
// ---------------------------------------------------------------------------
__global__ void __launch_bounds__(192)
tensor_encode(const float* __restrict__ xin,
              const float* __restrict__ planesT,   // (3,300,300,48)
              const float* __restrict__ linesT,    // (3,300,48)
              float* __restrict__ out) {           // (N,144)
    const int c6 = threadIdx.x;                     // 0..5
    const int p  = blockIdx.x * 32 + threadIdx.y;   // point id
    if (p >= NPTS) return;

    const float xs0 = fmaf(2.0f, xin[3 * p + 0], -1.0f);
    const float xs1 = fmaf(2.0f, xin[3 * p + 1], -1.0f);
    const float xs2 = fmaf(2.0f, xin[3 * p + 2], -1.0f);

    const float gxa[3] = { xs0, xs0, xs1 };
    const float gya[3] = { xs1, xs2, xs2 };
    const float gla[3] = { xs2, xs1, xs0 };

    int cell00[3], cell01[3], cell10[3], cell11[3], lo0[3], lo1[3];
    float ax[3], ay[3], al[3];
#pragma unroll
    for (int s = 0; s < 3; ++s) {
        int x0, x1, y0, y1, l0, l1;
        coord(gxa[s], x0, x1, ax[s]);
        coord(gya[s], y0, y1, ay[s]);
        coord(gla[s], l0, l1, al[s]);
        const int r0 = y0 * RES, r1 = y1 * RES;
        cell00[s] = (r0 + x0) * C4;
        cell01[s] = (r0 + x1) * C4;
        cell10[s] = (r1 + x0) * C4;
        cell11[s] = (r1 + x1) * C4;
        lo0[s] = l0 * C4;
        lo1[s] = l1 * C4;
    }

    const v4f* planesv = (const v4f*)planesT;
    const v4f* linesv  = (const v4f*)linesT;
    v4f* outv = (v4f*)out;
    const int outbase = p * 36;

    // get the gathered rows moving (global_prefetch_b8)
#pragma unroll
    for (int s = 0; s < 3; ++s) {
        const v4f* pb = planesv + s * (CELLS * C4);
        __builtin_prefetch((const void*)(pb + cell00[s] + c6), 0, 0);
        __builtin_prefetch((const void*)(pb + cell10[s] + c6), 0, 0);
    }

#pragma unroll
    for (int s = 0; s < 3; ++s) {
        const v4f* pb = planesv + s * (CELLS * C4);
        const v4f* lb = linesv + s * (RES * C4);

        const float w00 = (1.0f - ax[s]) * (1.0f - ay[s]);
        const float w01 = ax[s] * (1.0f - ay[s]);
        const float w10 = (1.0f - ax[s]) * ay[s];
        const float w11 = ax[s] * ay[s];

#pragma unroll
        for (int h = 0; h < 2; ++h) {
            const int o = h * 6 + c6;                   // float4 group 0..11
            v4f f00 = pb[cell00[s] + o];
            v4f f01 = pb[cell01[s] + o];
            v4f f10 = pb[cell10[s] + o];
            v4f f11 = pb[cell11[s] + o];
            v4f pf = f00 * w00 + f01 * w01 + f10 * w10 + f11 * w11;

            v4f l0 = lb[lo0[s] + o];
            v4f l1 = lb[lo1[s] + o];
            v4f lf = l0 + (l1 - l0) * al[s];            // lerp

            __builtin_nontemporal_store(pf * lf, &outv[outbase + s * C4 + o]);
        }
    }
}

extern "C" void kernel_launch(void* const* d_in, const int* in_sizes, int n_in,
                              void* d_out, int out_size, void* d_ws, size_t ws_size,
                              hipStream_t stream) {
    const float* xin = (const float*)d_in[0];   // (1M, 3)
    const float* pp  = (const float*)d_in[1];   // (3, 48, 300, 300)
    const float* lp  = (const float*)d_in[2];   // (3, 48, 300, 1)
    float* outp = (float*)d_out;                // (1M, 144)

    float* planesT = (float*)d_ws;              // 51.84 MB
    float* linesT  = planesT + (long)PLANE_F;   // 172.8 KB

    // 1) transpose planes via Tensor Data Mover: one block per (y, s)
    {
        dim3 blk(48, 6, 1);
        dim3 grid(RES, 3, 1);
        tpose_planes_tdm<<<grid, blk, 0, stream>>>(pp, planesT);
    }
    // 2) transpose lines
    {
        dim3 blk(320, 1, 1);
        dim3 grid(1, 3 * NC, 1);
        tpose_lines<<<grid, blk, 0, stream>>>(lp, linesT);
    }
    // 3) main encode: 32 points per block, (6,32) threads
    {
        dim3 blk(6, 32, 1);
        dim3 grid((NPTS + 31) / 32, 1, 1);
        tensor_encode<<<grid, blk, 0, stream>>>(xin, planesT, linesT, outp);
    }
}